// DropStripes_72507637891182
// MI455X (gfx1250) — compile-verified
//
#include <hip/hip_runtime.h>

// DropStripes: out[b,t,f] = x[b,t,f] * keep[b,f]
// keep[b,f] = !( any_s (f >= bgn[b,s] && f < bgn[b,s]+distance[b,s]) )
// B=64, T=2000, F=512, STRIPES=2. Pure streaming op: 524 MB @ 23.3 TB/s ~ 22.5 us.
//
// CDNA5 path: 8-deep pipeline of GLOBAL_LOAD_ASYNC_TO_LDS_B128 (all issued
// back-to-back for max memory-level parallelism), drained with explicit
// s_wait_asynccnt, masked in VGPRs, written back with non-temporal b128 stores.

typedef float f4v __attribute__((ext_vector_type(4)));
typedef int v4i __attribute__((vector_size(16))); // matches builtin prototype

#define AS1 __attribute__((address_space(1)))
#define AS3 __attribute__((address_space(3)))

#ifndef __has_builtin
#define __has_builtin(x) 0
#endif

#if __has_builtin(__builtin_amdgcn_global_load_async_to_lds_b128)
#define USE_ASYNC_LDS 1
#else
#define USE_ASYNC_LDS 0
#endif

#define TPB 256
#define STAGES 8
#define BLOCK_F4 (TPB * STAGES) // 2048 float4 (32 KB) per block

__global__ __launch_bounds__(TPB) void drop_stripes_kernel(
    const f4v* __restrict__ x,
    const int* __restrict__ bgn,
    const int* __restrict__ dist,
    f4v* __restrict__ out,
    int f4_per_batch) {
  const int b = blockIdx.y;                 // uniform -> scalar loads below
  const unsigned t = threadIdx.x;
  const size_t base =
      (size_t)b * (size_t)f4_per_batch + (size_t)blockIdx.x * BLOCK_F4;

  // Stripe params: uniform per block -> s_load_b32
  const int s0 = bgn[2 * b + 0];
  const int s1 = bgn[2 * b + 1];
  const unsigned d0 = (unsigned)dist[2 * b + 0];
  const unsigned d1 = (unsigned)dist[2 * b + 1];

  // F = 512 floats = 128 float4 per row; STAGES*TPB is a multiple of 128,
  // so this thread's freq position is identical across all stages.
  const int f = (int)(t & 127u) * 4;
  f4v m;
#pragma unroll
  for (int j = 0; j < 4; ++j) {
    const int fi = f + j;
    // one unsigned range-check per stripe: fi in [s, s+d) <=> (uint)(fi-s) < d
    const bool drop =
        ((unsigned)(fi - s0) < d0) || ((unsigned)(fi - s1) < d1);
    m[j] = drop ? 0.0f : 1.0f;
  }

  const f4v* __restrict__ src = x + base;
  f4v* __restrict__ dst = out + base;

#if USE_ASYNC_LDS
  // Each lane stages its own 16B slot; each buffer is written once and read
  // once, so per-wave ASYNCcnt ordering (loads complete in order) is the only
  // synchronization needed — no barriers, no WAR hazards.
  __shared__ f4v buf[STAGES][TPB];
#pragma unroll
  for (int s = 0; s < STAGES; ++s) {
    __builtin_amdgcn_global_load_async_to_lds_b128(
        (AS1 v4i*)(void*)(src + s * TPB + t),
        (AS3 v4i*)(void*)&buf[s][t], 0, 0);
  }
#pragma unroll
  for (int s = 0; s < STAGES; ++s) {
    // stage s is done once at most (STAGES-1-s) newer loads remain in flight
    asm volatile("s_wait_asynccnt %0" ::"i"(STAGES - 1 - s) : "memory");
    f4v v = buf[s][t];   // ds_load_b128
    v *= m;
    __builtin_nontemporal_store(v, dst + s * TPB + t); // b128 store, TH=NT
  }
#else
  // Fallback: direct non-temporal 128-bit streaming, fully unrolled (8-deep).
#pragma unroll
  for (int s = 0; s < STAGES; ++s) {
    f4v v = __builtin_nontemporal_load(src + s * TPB + t);
    v *= m;
    __builtin_nontemporal_store(v, dst + s * TPB + t);
  }
#endif
}

extern "C" void kernel_launch(void* const* d_in, const int* in_sizes, int n_in,
                              void* d_out, int out_size, void* d_ws,
                              size_t ws_size, hipStream_t stream) {
  const f4v* x = (const f4v*)d_in[0];
  const int* bgn = (const int*)d_in[1];
  const int* dist = (const int*)d_in[2];
  f4v* out = (f4v*)d_out;

  const int B = in_sizes[1] / 2;                  // STRIPES = 2
  const int f4_per_batch = (in_sizes[0] / B) / 4; // 256000 for (2000,512)
  const int bx = f4_per_batch / BLOCK_F4;         // 125
  dim3 grid(bx, B);
  drop_stripes_kernel<<<grid, dim3(TPB), 0, stream>>>(x, bgn, dist, out,
                                                      f4_per_batch);
}